// DataGNNPositionalEncodings_8143257994109
// MI455X (gfx1250) — compile-verified
//
#include <hip/hip_runtime.h>

typedef __attribute__((ext_vector_type(2))) float v2f;
typedef __attribute__((ext_vector_type(8))) float v8f;

#define N_FEAT 128

// ---------------------------------------------------------------------------
// Degree / norm precompute
// ---------------------------------------------------------------------------
__global__ void deg_kernel(const long long* __restrict__ dst,
                           float* __restrict__ deg, int n_edges) {
  int i = blockIdx.x * blockDim.x + threadIdx.x;
  if (i < n_edges) unsafeAtomicAdd(&deg[dst[i]], 1.0f);
}

__global__ void dis_kernel(float* __restrict__ deg, int n) {
  int i = blockIdx.x * blockDim.x + threadIdx.x;
  if (i < n) {
    float d = deg[i];
    deg[i] = (d > 0.0f) ? rsqrtf(d) : 0.0f;   // deg>0 -> integer >= 1
  }
}

__global__ void norm_kernel(const long long* __restrict__ src,
                            const long long* __restrict__ dst,
                            const float* __restrict__ dis,
                            float* __restrict__ nrm, int n_edges) {
  int i = blockIdx.x * blockDim.x + threadIdx.x;
  if (i < n_edges) nrm[i] = dis[src[i]] * dis[dst[i]];
}

// ---------------------------------------------------------------------------
// Sparse propagation: out[dst] += h[src] * norm  (one edge per wave32,
// one float4 per lane => 128 floats/row; f32 HW atomics resolve in L2)
// ---------------------------------------------------------------------------
__global__ __launch_bounds__(256) void prop_kernel(
    const float* __restrict__ h, const long long* __restrict__ src,
    const long long* __restrict__ dst, const float* __restrict__ nrm,
    float* __restrict__ out, int n_edges) {
  const int lane = threadIdx.x & 31;
  int w = (blockIdx.x * blockDim.x + threadIdx.x) >> 5;
  const int nw = (gridDim.x * blockDim.x) >> 5;
  for (int e = w; e < n_edges; e += nw) {
    const long long s = src[e];
    const long long d = dst[e];
    const float sc = nrm[e];
    float4 v = *(const float4*)(h + (size_t)s * N_FEAT + lane * 4);
    float* op = out + (size_t)d * N_FEAT + lane * 4;
    unsafeAtomicAdd(op + 0, v.x * sc);
    unsafeAtomicAdd(op + 1, v.y * sc);
    unsafeAtomicAdd(op + 2, v.z * sc);
    unsafeAtomicAdd(op + 3, v.w * sc);
  }
}

// ---------------------------------------------------------------------------
// LDS B-tile layout: K-pair interleaved + XOR-32 swizzle.
//   element (i, o) -> dword (i/2)*256 + ((o*2 + (i&1)) ^ ((i/2 & 1) * 32))
// Load side: one aligned ds_load_b64 per fragment, conflict-free banks.
// ---------------------------------------------------------------------------
__device__ __forceinline__ void stage_row(float* ldsB, int irow, int o,
                                          float4 v) {
  const int p = irow >> 1;
  const int lsb = irow & 1;
  const int sw = (p & 1) << 5;
  const int base = p << 8;
  ldsB[base + ((((o + 0) << 1) | lsb) ^ sw)] = v.x;
  ldsB[base + ((((o + 1) << 1) | lsb) ^ sw)] = v.y;
  ldsB[base + ((((o + 2) << 1) | lsb) ^ sw)] = v.z;
  ldsB[base + ((((o + 3) << 1) | lsb) ^ sw)] = v.w;
}

// ---------------------------------------------------------------------------
// Fused TAGConv GEMM: y = sum_{k=0..3} h_k @ W[k] + b, with epilogue.
// Block = 8 waves; each wave owns 16 rows x 128 cols held in 8 f32 WMMA accs.
// EPI: 0 = none, 1 = LayerNorm+ReLU, 2 = ReLU + skip
// ---------------------------------------------------------------------------
template <int EPI>
__global__ __launch_bounds__(256) void tag_gemm_kernel(
    const float* __restrict__ h0, const float* __restrict__ h1,
    const float* __restrict__ h2, const float* __restrict__ h3,
    const float* __restrict__ W,      // [4,128,128] row-major (in, out)
    const float* __restrict__ bias,   // [128]
    const float* __restrict__ skip,   // EPI==2
    const float* __restrict__ ln_g, const float* __restrict__ ln_b,
    float* __restrict__ y, int n_nodes) {
  __shared__ float ldsB[N_FEAT * N_FEAT];  // 64 KB: one hop's weights
  const int lane = threadIdx.x & 31;
  const int wave = threadIdx.x >> 5;
  const int half = lane >> 4;   // lane 0-15 vs 16-31
  const int l16  = lane & 15;
  const int row0 = blockIdx.x * 128 + wave * 16;

  int arow = row0 + l16;                      // A-fragment row (clamped tail)
  if (arow >= n_nodes) arow = n_nodes - 1;
  const size_t aoff = (size_t)arow * N_FEAT + half * 2;

  v8f acc[8];
#pragma unroll
  for (int j = 0; j < 8; ++j) {
    float bv = bias[j * 16 + l16];
    v8f t = {bv, bv, bv, bv, bv, bv, bv, bv};
    acc[j] = t;
  }

  const float* hops[4] = {h0, h1, h2, h3};
#pragma unroll
  for (int hop = 0; hop < 4; ++hop) {
    __syncthreads();
    {  // stage W[hop] into LDS (4096 float4 reads, swizzled scatter stores)
      const float4* sp = (const float4*)(W + (size_t)hop * N_FEAT * N_FEAT);
#pragma unroll
      for (int i = 0; i < 16; ++i) {
        int idx = i * 256 + threadIdx.x;     // float4 index
        float4 v = sp[idx];
        int flat = idx * 4;                  // = irow*128 + o
        stage_row(ldsB, flat >> 7, flat & 127, v);
      }
    }
    __syncthreads();

    const float* ap = hops[hop] + aoff;
    // per-lane LDS base: pair p = s*2+half; (p&1)==half folds the swizzle
    const int lbase = (half << 8) + (l16 << 1);
    for (int s = 0; s < 32; ++s) {           // K = 128 in steps of 4
      float2 af = *(const float2*)(ap + s * 4);
      v2f a; a.x = af.x; a.y = af.y;
      const int sbase = (s << 9) + lbase;    // (s*2+half)*256 + l16*2
      v2f bf[8];
#pragma unroll
      for (int j = 0; j < 8; ++j)
        bf[j] = *(const v2f*)(ldsB + sbase + ((j ^ half) << 5));
#pragma unroll
      for (int j = 0; j < 8; ++j)
        acc[j] = __builtin_amdgcn_wmma_f32_16x16x4_f32(
            false, a, false, bf[j], (short)0, acc[j], false, false);
    }
  }

  // ---- epilogue ----
  float meanv[8], rstd[8];
  if (EPI == 1) {
#pragma unroll
    for (int r = 0; r < 8; ++r) {            // row m = r + 8*half
      float s1 = 0.0f, s2 = 0.0f;
#pragma unroll
      for (int j = 0; j < 8; ++j) {
        float v = acc[j][r];
        s1 += v; s2 += v * v;
      }
#pragma unroll
      for (int m = 1; m <= 8; m <<= 1) {     // reduce across the 16-lane half
        s1 += __shfl_xor(s1, m, 32);
        s2 += __shfl_xor(s2, m, 32);
      }
      float mu = s1 * (1.0f / 128.0f);
      float var = s2 * (1.0f / 128.0f) - mu * mu;
      meanv[r] = mu;
      rstd[r] = rsqrtf(var + 1e-5f);
    }
  }

#pragma unroll
  for (int j = 0; j < 8; ++j) {
    const int n = j * 16 + l16;
    float g = 0.0f, be = 0.0f;
    if (EPI == 1) { g = ln_g[n]; be = ln_b[n]; }
#pragma unroll
    for (int r = 0; r < 8; ++r) {
      int gm = row0 + half * 8 + r;
      float v = acc[j][r];
      if (EPI == 1) {
        v = (v - meanv[r]) * rstd[r] * g + be;
        v = fmaxf(v, 0.0f);
      } else if (EPI == 2) {
        v = fmaxf(v, 0.0f);
        int sr = (gm < n_nodes) ? gm : (n_nodes - 1);
        v += skip[(size_t)sr * N_FEAT + n];
      }
      if (gm < n_nodes) y[(size_t)gm * N_FEAT + n] = v;
    }
  }
}

// ---------------------------------------------------------------------------
// Output projection: out = A(Nx128) @ Wp(128x256) + bias, 2 column chunks
// ---------------------------------------------------------------------------
__global__ __launch_bounds__(256) void proj_gemm_kernel(
    const float* __restrict__ A, const float* __restrict__ Wp,
    const float* __restrict__ bias, float* __restrict__ y, int n_nodes) {
  __shared__ float ldsB[N_FEAT * N_FEAT];
  const int lane = threadIdx.x & 31;
  const int wave = threadIdx.x >> 5;
  const int half = lane >> 4;
  const int l16  = lane & 15;
  const int row0 = blockIdx.x * 128 + wave * 16;

  int arow = row0 + l16;
  if (arow >= n_nodes) arow = n_nodes - 1;
  const size_t aoff = (size_t)arow * N_FEAT + half * 2;

#pragma unroll
  for (int ch = 0; ch < 2; ++ch) {
    __syncthreads();
    {  // stage Wp[:, ch*128 : ch*128+128] (row stride 256 in global)
#pragma unroll
      for (int i = 0; i < 16; ++i) {
        int flat = i * 1024 + threadIdx.x * 4;   // float index 0..16380
        int irow = flat >> 7, o = flat & 127;
        float4 v =
            *(const float4*)(Wp + (size_t)irow * 256 + ch * 128 + o);
        stage_row(ldsB, irow, o, v);
      }
    }
    __syncthreads();

    v8f acc[8];
#pragma unroll
    for (int j = 0; j < 8; ++j) {
      float bv = bias[ch * 128 + j * 16 + l16];
      v8f t = {bv, bv, bv, bv, bv, bv, bv, bv};
      acc[j] = t;
    }

    const float* ap = A + aoff;
    const int lbase = (half << 8) + (l16 << 1);
    for (int s = 0; s < 32; ++s) {
      float2 af = *(const float2*)(ap + s * 4);
      v2f a; a.x = af.x; a.y = af.y;
      const int sbase = (s << 9) + lbase;
      v2f bf[8];
#pragma unroll
      for (int j = 0; j < 8; ++j)
        bf[j] = *(const v2f*)(ldsB + sbase + ((j ^ half) << 5));
#pragma unroll
      for (int j = 0; j < 8; ++j)
        acc[j] = __builtin_amdgcn_wmma_f32_16x16x4_f32(
            false, a, false, bf[j], (short)0, acc[j], false, false);
    }

#pragma unroll
    for (int j = 0; j < 8; ++j) {
      const int n = ch * 128 + j * 16 + l16;
#pragma unroll
      for (int r = 0; r < 8; ++r) {
        int gm = row0 + half * 8 + r;
        if (gm < n_nodes) y[(size_t)gm * 256 + n] = acc[j][r];
      }
    }
  }
}

// ---------------------------------------------------------------------------
extern "C" void kernel_launch(void* const* d_in, const int* in_sizes, int n_in,
                              void* d_out, int out_size, void* d_ws,
                              size_t ws_size, hipStream_t stream) {
  const float* x      = (const float*)d_in[0];
  const long long* ei = (const long long*)d_in[1];
  const float* w0 = (const float*)d_in[2];
  const float* b0 = (const float*)d_in[3];
  const float* w1 = (const float*)d_in[4];
  const float* b1 = (const float*)d_in[5];
  const float* w2 = (const float*)d_in[6];
  const float* b2 = (const float*)d_in[7];
  const float* lg = (const float*)d_in[8];
  const float* lb = (const float*)d_in[9];
  const float* pw = (const float*)d_in[10];
  const float* pb = (const float*)d_in[11];
  float* out = (float*)d_out;

  const int n = in_sizes[0] / N_FEAT;   // 100000 nodes
  const int e = in_sizes[1] / 2;        // 1600000 edges
  const long long* esrc = ei;
  const long long* edst = ei + e;

  const size_t nodef = (size_t)n * N_FEAT;  // floats per node-feature buffer

  char* ws = (char*)d_ws;
  float* dis = (float*)ws; ws += (((size_t)n * 4) + 255) & ~(size_t)255;
  float* nrm = (float*)ws; ws += (((size_t)e * 4) + 255) & ~(size_t)255;
  float* p1  = (float*)ws; ws += nodef * 4;
  float* Y0  = (float*)ws; ws += nodef * 4;
  float* Y1  = (float*)ws; ws += nodef * 4;
  // p2/p3 live inside d_out (exactly 2 * N*128 f32); final proj overwrites it.
  float* p2 = out;
  float* p3 = out + nodef;
  float* Y2 = p1;  // row-wise-safe alias: each wave reads its rows before storing

  const int TB = 256;
  const int eb = (e + TB - 1) / TB;
  const int nb = (n + TB - 1) / TB;
  const int gb = (n + 127) / 128;
  const int propb = 2048;

  // degree -> deg^{-1/2} -> per-edge norm
  hipMemsetAsync(dis, 0, (size_t)n * 4, stream);
  deg_kernel<<<eb, TB, 0, stream>>>(edst, dis, e);
  dis_kernel<<<nb, TB, 0, stream>>>(dis, n);
  norm_kernel<<<eb, TB, 0, stream>>>(esrc, edst, dis, nrm, e);

  auto prop = [&](const float* hs, float* hd) {
    hipMemsetAsync(hd, 0, nodef * 4, stream);
    prop_kernel<<<propb, TB, 0, stream>>>(hs, esrc, edst, nrm, hd, e);
  };

  // layer 0: conv0 -> LayerNorm -> ReLU
  prop(x, p1); prop(p1, p2); prop(p2, p3);
  tag_gemm_kernel<1><<<gb, TB, 0, stream>>>(x, p1, p2, p3, w0, b0, nullptr,
                                            lg, lb, Y0, n);
  // layer 1: conv1 -> ReLU -> +skip
  prop(Y0, p1); prop(p1, p2); prop(p2, p3);
  tag_gemm_kernel<2><<<gb, TB, 0, stream>>>(Y0, p1, p2, p3, w1, b1, Y0,
                                            lg, lb, Y1, n);
  // layer 2: conv2 (no epilogue)
  prop(Y1, p1); prop(p1, p2); prop(p2, p3);
  tag_gemm_kernel<0><<<gb, TB, 0, stream>>>(Y1, p1, p2, p3, w2, b2, nullptr,
                                            lg, lb, Y2, n);
  // projection 128 -> 256
  proj_gemm_kernel<<<gb, TB, 0, stream>>>(Y2, pw, pb, out, n);
}